// DenseCRFLoss_19507741458472
// MI455X (gfx1250) — compile-verified
//
#include <hip/hip_runtime.h>
#include <hip/hip_bf16.h>

typedef __attribute__((ext_vector_type(2))) float v2f;
typedef __attribute__((ext_vector_type(8))) float v8f;

#define IMG_H   96
#define IMG_W   96
#define HW      (IMG_H * IMG_W)       // 9216
#define NTILES  (HW / 16)             // 576
#define NFOLD   (NTILES / 2)          // 288: block f covers i-tiles {f, 575-f}
#define NIMG    2
#define NCLS    2
// WEIGHT = 2e-9, loss = WEIGHT * (-sum / N)  ->  scale = -1e-9
#define OUT_SCALE (-1.0e-9f)

// gray scale: (255/15) * sqrt(3)  (3 identical rgb channels folded into one dim)
#define GSCALE  29.44486373f
#define XSCALE  0.01f                 // 1/SIGMA_XY

__global__ void densecrf_zero(float* out) { out[0] = 0.0f; }

__global__ __launch_bounds__(256)
void densecrf_wmma(const float* __restrict__ img,   // [N,1,H,W]
                   const float* __restrict__ seg,   // [N,K,H,W]
                   float* __restrict__ out)         // [1]
{
    __shared__ float sqI[2][16];
    __shared__ float wsum[8];

    const int tid  = threadIdx.x;
    const int lane = tid & 31;
    // force scalar: waves then run uniform s_cbranch loops, EXEC stays all-1s
    const int wave = __builtin_amdgcn_readfirstlane(tid >> 5);
    const int bid  = blockIdx.x;          // 0 .. NIMG*NFOLD-1
    const int n    = bid / NFOLD;         // image index
    const int fold = bid % NFOLD;         // triangle fold index

    const float* __restrict__ imgn = img + n * HW;
    const float* __restrict__ s0p  = seg + (n * NCLS + 0) * HW;
    const float* __restrict__ s1p  = seg + (n * NCLS + 1) * HW;

    // warm the per-image working set (~110 KB) into cache before 577 reuse passes
    __builtin_prefetch(imgn + tid * (HW / 256), 0, 0);
    __builtin_prefetch(s0p  + tid * (HW / 256), 0, 0);
    __builtin_prefetch(s1p  + tid * (HW / 256), 0, 0);

    // ---- per-row squared feature norms for both i-tiles of this fold ----
    if (tid < 32) {
        const int p  = tid >> 4;                       // phase 0/1
        const int it = p ? (NTILES - 1 - fold) : fold;
        const int i  = it * 16 + (tid & 15);
        float fx = (float)(i % IMG_W) * XSCALE;
        float fy = (float)(i / IMG_W) * XSCALE;
        float g  = imgn[i] * GSCALE;
        sqI[p][tid & 15] = fx * fx + fy * fy + g * g;
    }
    __syncthreads();

    const int   m     = lane & 15;                 // row (A) / col (B) in tile
    const int   kh    = lane >> 4;                 // which K-half this lane supplies
    const float kmask = kh ? 0.0f : 1.0f;          // branchless lane masking

    // ---- A fragments (feature + seg) + row biases for both phases ----
    // A 16x4 f32 layout: lane l -> row l&15 ; VGPR v -> K = 2*(l>>4)+v
    v2f   af[2], as[2];
    float sqIr[2][8];
#pragma unroll
    for (int p = 0; p < 2; ++p) {
        const int it = p ? (NTILES - 1 - fold) : fold;
        const int ia = it * 16 + m;
        const float fxa = (float)(ia % IMG_W) * XSCALE;
        const float fya = (float)(ia / IMG_W) * XSCALE;
        const float ga  = imgn[ia] * GSCALE;
        af[p].x = kh ? ga : fxa;          // K = {0:fx, 1:fy, 2:sqrt3*g, 3:0}
        af[p].y = fya * kmask;
        as[p].x = s0p[ia] * kmask;        // K = {0:S0, 1:S1, 2:0, 3:0}
        as[p].y = s1p[ia] * kmask;
        // D layout: VGPR r -> row (r + 8*kh), col (lane&15)
#pragma unroll
        for (int r = 0; r < 8; ++r) sqIr[p][r] = sqI[p][r + 8 * kh];
    }

    float acc = 0.0f;

    // ---- upper-triangle sweep: jt in [it, 575], weight 2 off-diagonal ----
#pragma unroll
    for (int p = 0; p < 2; ++p) {
        const int it    = p ? (NTILES - 1 - fold) : fold;
        const int start = it + wave;                       // scalar
        const int cnt   = (start < NTILES) ? ((NTILES - start + 7) >> 3) : 0;
        // scalar trip count -> uniform s_cbranch loop; EXEC all-1s at WMMA
        for (int q = 0; q < cnt; ++q) {
            const int jt = start + (q << 3);
            const float wgt = (jt == it) ? 1.0f : 2.0f;    // scalar select
            const int j = jt * 16 + m;
            const float fxb = (float)(j % IMG_W) * XSCALE;
            const float fyb = (float)(j / IMG_W) * XSCALE;
            const float gb  = imgn[j] * GSCALE;
            const float sqJ = fxb * fxb + fyb * fyb + gb * gb;

            v2f bf, bs;   // B 4x16 mirrors A layout (unconditional loads + mask)
            bf.x = kh ? gb : fxb;
            bf.y = fyb * kmask;
            bs.x = s0p[j] * kmask;
            bs.y = s1p[j] * kmask;

            v8f c = {};
            // dot[i][j] = f_i . f_j  (K=4 exact fit)
            v8f dotm = __builtin_amdgcn_wmma_f32_16x16x4_f32(
                false, af[p], false, bf, (short)0, c, false, false);
            // gram[i][j] = S0_i*S0_j + S1_i*S1_j
            v8f gram = __builtin_amdgcn_wmma_f32_16x16x4_f32(
                false, as[p], false, bs, (short)0, c, false, false);

            float tacc = 0.0f;
#pragma unroll
            for (int r = 0; r < 8; ++r) {
                // Kmat = exp(dot - 0.5*(sq_i + sq_j))
                float e = __expf(dotm[r] - 0.5f * (sqIr[p][r] + sqJ));
                tacc = fmaf(e, gram[r], tacc);
            }
            acc = fmaf(wgt, tacc, acc);
        }
    }

    // ---- wave32 reduce, then block reduce, then one atomic ----
#pragma unroll
    for (int off = 16; off > 0; off >>= 1)
        acc += __shfl_xor(acc, off, 32);
    if (lane == 0) wsum[wave] = acc;
    __syncthreads();

    if (tid == 0) {
        float s = 0.0f;
#pragma unroll
        for (int w = 0; w < 8; ++w) s += wsum[w];
        atomicAdd(out, s * OUT_SCALE);
    }
}

extern "C" void kernel_launch(void* const* d_in, const int* in_sizes, int n_in,
                              void* d_out, int out_size, void* d_ws, size_t ws_size,
                              hipStream_t stream) {
    const float* images = (const float*)d_in[0];   // [2,1,96,96] f32
    const float* segs   = (const float*)d_in[1];   // [2,2,96,96] f32
    float* out = (float*)d_out;                    // [1] f32

    densecrf_zero<<<1, 1, 0, stream>>>(out);
    densecrf_wmma<<<NIMG * NFOLD, 256, 0, stream>>>(images, segs, out);
}